// SSIM_17772574670848
// MI455X (gfx1250) — compile-verified
//
#include <hip/hip_runtime.h>

// ---------------------------------------------------------------------------
// SSIM (7x7 uniform window) fused kernel for gfx1250 (MI455X).
// - Box filters realized as V_WMMA_F32_16X16X4_F32 against constant banded
//   ones/7 matrices (separable 7-tap pass == banded matmul), fp32 throughout.
// - Tile fill uses CDNA5 async global->LDS b128 copies (ASYNCcnt).
// ---------------------------------------------------------------------------

typedef float v2f __attribute__((ext_vector_type(2)));
typedef float v8f __attribute__((ext_vector_type(8)));

#define IMG_N     384
#define OUT_N     378          // valid outputs per dim (crop removes pad band)
#define TILE      64
#define NT        6            // 6*64 = 384 >= 378 tiles per dim
#define IN_STRIDE 84           // row pitch 336B (16B aligned) for b128 LDS stores
#define H_STRIDE  66           // H buffer pitch (bank spread)
#define COV_NORM  (49.0f / 48.0f)

// banded 7-tap matrix entry: band(k, j) = 1/7 iff j <= k <= j+6
__device__ __forceinline__ float band7(int k, int j) {
  return (k >= j && k <= j + 6) ? (1.0f / 7.0f) : 0.0f;
}

// ---------------------------- init workspace -------------------------------
__global__ void ssim_init_kernel(unsigned* maxbits, double* sum) {
  *maxbits = 0u;     // all gt values are >= 0
  *sum = 0.0;
}

// ---------------------------- max(gt) reduction ----------------------------
__global__ __launch_bounds__(256)
void ssim_max_kernel(const float* __restrict__ gt, int n, unsigned* __restrict__ maxbits) {
  __shared__ float red[256];
  int tid = threadIdx.x;
  float m = 0.0f;
  for (int i = blockIdx.x * blockDim.x + tid; i < n; i += gridDim.x * blockDim.x)
    m = fmaxf(m, gt[i]);
  red[tid] = m;
  __syncthreads();
  for (int off = 128; off > 0; off >>= 1) {
    if (tid < off) red[tid] = fmaxf(red[tid], red[tid + off]);
    __syncthreads();
  }
  if (tid == 0) atomicMax(maxbits, __float_as_uint(red[0]));
}

// ------------------------- fused SSIM tile kernel --------------------------
__global__ __launch_bounds__(256)
void ssim_tile_kernel(const float* __restrict__ gt, const float* __restrict__ pred,
                      const unsigned* __restrict__ maxbits, double* __restrict__ sum_out) {
  extern __shared__ __align__(16) float smem[];
  float* sGT  = smem;                                      // 80 x 84
  float* sPR  = smem + 80 * IN_STRIDE;                     // 80 x 84
  float* sH   = smem + 2 * 80 * IN_STRIDE;                 // 80 x 66 (per-map reuse)
  float* sRed = smem + 2 * 80 * IN_STRIDE + 80 * H_STRIDE; // 256

  const int tid  = threadIdx.x;
  const int lane = tid & 31;
  // Wave id as a true SGPR: keeps tile loops scalar so EXEC is provably all-1s
  // at every WMMA (ISA requirement), instead of a runtime-uniform masked loop.
  const int wave = __builtin_amdgcn_readfirstlane(tid >> 5);   // 8 waves
  const int lo   = lane & 15;
  const int hi   = lane >> 4;     // lane half (0/1)

  const int b   = blockIdx.x;
  const int img = b / (NT * NT);
  const int t2  = b % (NT * NT);
  const int Y0  = (t2 / NT) * TILE;
  const int X0  = (t2 % NT) * TILE;
  const float* gimg = gt   + (size_t)img * IMG_N * IMG_N;
  const float* pimg = pred + (size_t)img * IMG_N * IMG_N;

  // Zero-fill input tiles (covers out-of-image halo and the 16-align padding
  // rows/cols: those are only ever multiplied by zero band entries, but must
  // be finite).
  for (int i = tid; i < 2 * 80 * IN_STRIDE; i += 256) smem[i] = 0.0f;
  __syncthreads();

  // Load the 70x72 input region (64 outputs + 6 halo, rounded to float4) with
  // async global->LDS b128 copies; image-edge fragments fall back to guarded
  // scalar loads. Both global and LDS addresses are 16B aligned by layout.
  for (int i = tid; i < 70 * 18; i += 256) {
    int r  = i / 18;
    int c  = 4 * (i % 18);             // col 0..68 step 4 (cols 70..71 spill)
    int gy = Y0 + r, gx = X0 + c;
    if (gy < IMG_N) {
      if (gx + 3 < IMG_N) {
        unsigned ldsG = (unsigned)(uintptr_t)&sGT[r * IN_STRIDE + c];
        unsigned ldsP = (unsigned)(uintptr_t)&sPR[r * IN_STRIDE + c];
        unsigned long long ga = (unsigned long long)(uintptr_t)(gimg + gy * IMG_N + gx);
        unsigned long long pa = (unsigned long long)(uintptr_t)(pimg + gy * IMG_N + gx);
        asm volatile("global_load_async_to_lds_b128 %0, %1, off"
                     :: "v"(ldsG), "v"(ga) : "memory");
        asm volatile("global_load_async_to_lds_b128 %0, %1, off"
                     :: "v"(ldsP), "v"(pa) : "memory");
      } else {
        for (int e = 0; e < 4; ++e) {
          if (gx + e < IMG_N) {
            sGT[r * IN_STRIDE + c + e] = gimg[gy * IMG_N + gx + e];
            sPR[r * IN_STRIDE + c + e] = pimg[gy * IMG_N + gx + e];
          }
        }
      }
    }
  }
  asm volatile("s_wait_asynccnt 0x0" ::: "memory");   // async LDS writes landed
  __syncthreads();

  // Constant banded fragments. For K-chunk kk, this lane holds:
  //  as B (4x16):  M[k0][lo], M[k0+1][lo]   (k0 = 4*kk + 2*hi)
  //  as A (16x4):  N[lo][k0], N[lo][k0+1]
  // and M[k][j] == N[j][k] == band7(k, j), so one fragment serves both passes.
  v2f bandfrag[6];
#pragma unroll
  for (int kk = 0; kk < 6; ++kk) {
    int k0 = 4 * kk + 2 * hi;
    bandfrag[kk].x = band7(k0, lo);
    bandfrag[kk].y = band7(k0 + 1, lo);
  }

  // Vertical-pass accumulators: 5 maps x 2 subtiles per wave (16x16 each).
  v8f acc[5][2];
#pragma unroll
  for (int m = 0; m < 5; ++m)
#pragma unroll
    for (int t = 0; t < 2; ++t) acc[m][t] = (v8f){0, 0, 0, 0, 0, 0, 0, 0};

  for (int m = 0; m < 5; ++m) {   // maps: x, y, x*x, y*y, x*y
    // ---- horizontal pass: 5 row-blocks x 4 col-subtiles = 20 WMMA tiles ----
    for (int th = wave; th < 20; th += 8) {       // scalar loop (wave uniform)
      int r = th >> 2, c = th & 3;
      v8f h = (v8f){0, 0, 0, 0, 0, 0, 0, 0};
#pragma unroll
      for (int kk = 0; kk < 6; ++kk) {
        int row = 16 * r + lo;
        int col = 16 * c + 4 * kk + 2 * hi;       // even -> ds_load_b64
        float g0 = sGT[row * IN_STRIDE + col];
        float g1 = sGT[row * IN_STRIDE + col + 1];
        float p0 = sPR[row * IN_STRIDE + col];
        float p1 = sPR[row * IN_STRIDE + col + 1];
        v2f a;
        if      (m == 0) { a.x = g0;      a.y = g1;      }
        else if (m == 1) { a.x = p0;      a.y = p1;      }
        else if (m == 2) { a.x = g0 * g0; a.y = g1 * g1; }
        else if (m == 3) { a.x = p0 * p0; a.y = p1 * p1; }
        else             { a.x = g0 * p0; a.y = g1 * p1; }
        // H(16x16) += A(16x4) x M(4x16)
        h = __builtin_amdgcn_wmma_f32_16x16x4_f32(false, a, false, bandfrag[kk],
                                                  (short)0, h, false, false);
      }
#pragma unroll
      for (int v = 0; v < 8; ++v)
        sH[(16 * r + v + 8 * hi) * H_STRIDE + 16 * c + lo] = h[v];
    }
    __syncthreads();

    // ---- vertical pass: each wave owns 2 of the 16 output subtiles ----
#pragma unroll
    for (int t = 0; t < 2; ++t) {
      int s = 2 * wave + t;                       // scalar
      int rp = s >> 2, c = s & 3;
      v8f d = acc[m][t];
#pragma unroll
      for (int kk = 0; kk < 6; ++kk) {
        int k0 = 4 * kk + 2 * hi;
        v2f bb;
        bb.x = sH[(16 * rp + k0    ) * H_STRIDE + 16 * c + lo];
        bb.y = sH[(16 * rp + k0 + 1) * H_STRIDE + 16 * c + lo];
        // D(16x16) += N(16x4) x H(4x16)
        d = __builtin_amdgcn_wmma_f32_16x16x4_f32(false, bandfrag[kk], false, bb,
                                                  (short)0, d, false, false);
      }
      acc[m][t] = d;
    }
    __syncthreads();   // before next map overwrites sH
  }

  // ---- pointwise SSIM + masked reduction ----
  const float dr = __uint_as_float(*maxbits);
  const float C1 = (0.01f * dr) * (0.01f * dr);
  const float C2 = (0.03f * dr) * (0.03f * dr);

  float psum = 0.0f;
#pragma unroll
  for (int t = 0; t < 2; ++t) {
    int s = 2 * wave + t;
    int rp = s >> 2, c = s & 3;
#pragma unroll
    for (int v = 0; v < 8; ++v) {
      float mx  = acc[0][t][v], my  = acc[1][t][v];
      float exx = acc[2][t][v], eyy = acc[3][t][v], exy = acc[4][t][v];
      float vx  = COV_NORM * (exx - mx * mx);
      float vy  = COV_NORM * (eyy - my * my);
      float vxy = COV_NORM * (exy - mx * my);
      float num = (2.0f * mx * my + C1) * (2.0f * vxy + C2);
      float den = (mx * mx + my * my + C1) * (vx + vy + C2);
      float sv  = num / den;
      int oy = Y0 + 16 * rp + v + 8 * hi;
      int ox = X0 + 16 * c + lo;
      psum += (oy < OUT_N && ox < OUT_N) ? sv : 0.0f;
    }
  }
  sRed[tid] = psum;
  __syncthreads();
  for (int off = 128; off > 0; off >>= 1) {
    if (tid < off) sRed[tid] += sRed[tid + off];
    __syncthreads();
  }
  if (tid == 0) atomicAdd(sum_out, (double)sRed[0]);
}

// ------------------------------- finalize ----------------------------------
__global__ void ssim_finalize_kernel(const double* __restrict__ sum, float* __restrict__ out) {
  out[0] = (float)(*sum / (double)(64LL * OUT_N * OUT_N));
}

// ------------------------------- launcher ----------------------------------
extern "C" void kernel_launch(void* const* d_in, const int* in_sizes, int n_in,
                              void* d_out, int out_size, void* d_ws, size_t ws_size,
                              hipStream_t stream) {
  const float* gt   = (const float*)d_in[0];
  const float* pred = (const float*)d_in[1];
  // d_in[2] is the window (uniform ones/49) — folded into the banded matrices.

  unsigned* wsmax = (unsigned*)d_ws;                 // [0..3]  max(gt) bits
  double*   wssum = (double*)((char*)d_ws + 8);      // [8..15] SSIM sum
  float*    out   = (float*)d_out;

  const int n = in_sizes[0];                          // 64*384*384

  ssim_init_kernel<<<1, 1, 0, stream>>>(wsmax, wssum);
  ssim_max_kernel<<<2048, 256, 0, stream>>>(gt, n, wsmax);

  const size_t shmem =
      (size_t)(2 * 80 * IN_STRIDE + 80 * H_STRIDE + 256) * sizeof(float); // ~75.9 KB
  ssim_tile_kernel<<<64 * NT * NT, 256, shmem, stream>>>(gt, pred, wsmax, wssum);

  ssim_finalize_kernel<<<1, 1, 0, stream>>>(wssum, out);
}